// RenderingKaolinMulti_10960756539822
// MI455X (gfx1250) — compile-verified
//
#include <hip/hip_runtime.h>
#include <math.h>

#define HH 128
#define WW 128
#define NFRAMES 4
#define NFACES 320
#define NVERTS 162
#define FS 32               // face-setup stride in floats (128 bytes)
#define NCHUNK (NFACES/16)  // 20 chunks of 16 faces
#define SIGMAINV_C 7000.0f
#define CAMD 2.0f

// Face-setup record layout (FS floats):
//  [0..3]  : c1x, c1y, c1c, 0      (w1 affine coeffs, padded for b64 A-load)
//  [4..7]  : c2x, c2y, c2c, 0      (w2 affine coeffs)
//  [8..11] : zx,  zy,  zc,  0      (z   affine coeffs)
//  [12]    : valid flag (0/1)
//  [13..27]: 3 edges x {p0x, p0y, ex, ey, invL2}

typedef float v2f __attribute__((ext_vector_type(2)));
typedef float v8f __attribute__((ext_vector_type(8)));

__device__ __forceinline__ void aa_from_quat(const float q[4], float aa[3]) {
  float q1 = q[1], q2 = q[2], q3 = q[3];
  float sin_sq = q1*q1 + q2*q2 + q3*q3;
  float sin_t = sqrtf(sin_sq + 1e-20f);
  float cos_t = q[0];
  float two_theta = 2.0f * ((cos_t < 0.0f) ? atan2f(-sin_t, -cos_t)
                                           : atan2f(sin_t, cos_t));
  float k = (sin_sq > 1e-12f) ? (two_theta / sin_t) : 2.0f;
  aa[0] = q1*k; aa[1] = q2*k; aa[2] = q3*k;
}

__device__ __forceinline__ void rot_from_aa(const float aa[3], float R[9]) {
  float th = sqrtf(aa[0]*aa[0] + aa[1]*aa[1] + aa[2]*aa[2] + 1e-12f);
  float kx = aa[0]/th, ky = aa[1]/th, kz = aa[2]/th;
  float s = sinf(th), c = cosf(th), oc = 1.0f - c;
  R[0] = 1.0f + oc*(-(ky*ky + kz*kz));
  R[1] = s*(-kz) + oc*(kx*ky);
  R[2] = s*( ky) + oc*(kx*kz);
  R[3] = s*( kz) + oc*(kx*ky);
  R[4] = 1.0f + oc*(-(kx*kx + kz*kz));
  R[5] = s*(-kx) + oc*(ky*kz);
  R[6] = s*(-ky) + oc*(kx*kz);
  R[7] = s*( kx) + oc*(ky*kz);
  R[8] = 1.0f + oc*(-(kx*kx + ky*ky));
}

__device__ __forceinline__ void matmul3(const float A[9], const float B[9], float C[9]) {
  for (int i = 0; i < 3; i++)
    for (int j = 0; j < 3; j++)
      C[i*3+j] = A[i*3+0]*B[0*3+j] + A[i*3+1]*B[1*3+j] + A[i*3+2]*B[2*3+j];
}

// 4x async b128: copy this lane's 64 contiguous bytes global -> LDS (ASYNCcnt)
__device__ __forceinline__ void async_stage(const float* g, unsigned ldsoff) {
  asm volatile(
      "global_load_async_to_lds_b128 %0, %1, off\n\t"
      "global_load_async_to_lds_b128 %0, %1, off offset:16\n\t"
      "global_load_async_to_lds_b128 %0, %1, off offset:32\n\t"
      "global_load_async_to_lds_b128 %0, %1, off offset:48"
      :: "v"(ldsoff), "v"(g) : "memory");
}

// ---------------- Kernel 1: per-frame face setup ----------------
__global__ void setup_kernel(const float* __restrict__ vertices,
                             const float* __restrict__ translation,
                             const float* __restrict__ quaternion,
                             const float* __restrict__ expv_p,
                             const int*   __restrict__ faces,
                             float* __restrict__ fsetup) {
  __shared__ float sV[NVERTS*3];
  __shared__ float sR[9];
  __shared__ float sT[3];
  const int t = blockIdx.x;
  const int tid = threadIdx.x;

  if (tid == 0) {
    float expv = expv_p[0];
    float q0[4], q1[4];
    for (int i = 0; i < 4; i++) { q0[i] = quaternion[i]; q1[i] = quaternion[4+i]; }
    float aa0[3]; aa_from_quat(q0, aa0);
    float rot[9]; rot_from_aa(aa0, rot);
    float ang[3]; aa_from_quat(q1, ang);                 // ROT_DIVIDE == 1
    float aas[3] = { expv*ang[0]/4.0f, expv*ang[1]/4.0f, expv*ang[2]/4.0f };
    float rstep[9]; rot_from_aa(aas, rstep);
    float Rt[9]; for (int i = 0; i < 9; i++) Rt[i] = rot[i];
    float tmp[9];
    for (int i = 0; i < t; i++) { matmul3(Rt, rstep, tmp); for (int j = 0; j < 9; j++) Rt[j] = tmp[j]; }
    float tif  = (float)((double)t / 3.0);               // ti in linspace(0,1,4)
    float texp = tif * expv;                              // time_before = 0
    for (int i = 0; i < 9; i++) sR[i] = Rt[i];
    for (int c = 0; c < 3; c++) sT[c] = translation[c] + texp * translation[3+c];
  }
  __syncthreads();

  for (int v = tid; v < NVERTS; v += blockDim.x) {
    float x = vertices[v*3], y = vertices[v*3+1], z = vertices[v*3+2];
    sV[v*3+0] = sR[0]*x + sR[1]*y + sR[2]*z + sT[0];
    sV[v*3+1] = sR[3]*x + sR[4]*y + sR[5]*z + sT[1];
    sV[v*3+2] = sR[6]*x + sR[7]*y + sR[8]*z + sT[2] - CAMD;   // camera space
  }
  __syncthreads();

  const float FOCAL = (float)(1.0 / tan(1.57 / 4.0));
  for (int f = tid; f < NFACES; f += blockDim.x) {
    int i0 = faces[f*3+0], i1 = faces[f*3+1], i2 = faces[f*3+2];
    float a0 = sV[i0*3], a1 = sV[i0*3+1], a2 = sV[i0*3+2];
    float b0 = sV[i1*3], b1 = sV[i1*3+1], b2 = sV[i1*3+2];
    float c0 = sV[i2*3], c1 = sV[i2*3+1], c2 = sV[i2*3+2];
    // z of cross(b-a, c-a); positive normalization preserves sign
    float nz = (b0-a0)*(c1-a1) - (b1-a1)*(c0-a0);
    // projections
    float pax = FOCAL*a0/(-a2 + 1e-10f), pay = FOCAL*a1/(-a2 + 1e-10f);
    float pbx = FOCAL*b0/(-b2 + 1e-10f), pby = FOCAL*b1/(-b2 + 1e-10f);
    float pcx = FOCAL*c0/(-c2 + 1e-10f), pcy = FOCAL*c1/(-c2 + 1e-10f);
    float E0x = pbx-pax, E0y = pby-pay;
    float E1x = pcx-pax, E1y = pcy-pay;
    float den = E0x*E1y - E0y*E1x;
    bool  vden = fabsf(den) > 1e-10f;
    float inv  = vden ? 1.0f/den : 0.0f;
    bool  valid = vden && (nz > 0.0f);
    // w1 = (dx*E1y - dy*E1x)*inv ; w2 = (E0x*dy - E0y*dx)*inv  (dx=px-pax, dy=py-pay)
    float c1x =  E1y*inv, c1y = -E1x*inv, c1c = (pay*E1x - pax*E1y)*inv;
    float c2x = -E0y*inv, c2y =  E0x*inv, c2c = (E0y*pax - E0x*pay)*inv;
    float dz1 = b2 - a2, dz2 = c2 - a2;
    float zx = c1x*dz1 + c2x*dz2;
    float zy = c1y*dz1 + c2y*dz2;
    float zc = a2 + c1c*dz1 + c2c*dz2;

    float* o = fsetup + (size_t)(t*NFACES + f)*FS;
    o[0]=c1x; o[1]=c1y; o[2]=c1c;  o[3]=0.0f;
    o[4]=c2x; o[5]=c2y; o[6]=c2c;  o[7]=0.0f;
    o[8]=zx;  o[9]=zy;  o[10]=zc;  o[11]=0.0f;
    o[12] = valid ? 1.0f : 0.0f;
    float ppx[3] = {pax, pbx, pcx}, ppy[3] = {pay, pby, pcy};
    for (int e = 0; e < 3; e++) {
      int e2 = (e+1) % 3;
      float ex = ppx[e2]-ppx[e], ey = ppy[e2]-ppy[e];
      float L2 = ex*ex + ey*ey + 1e-12f;
      o[13+e*5+0] = ppx[e]; o[13+e*5+1] = ppy[e];
      o[13+e*5+2] = ex;     o[13+e*5+3] = ey;    o[13+e*5+4] = 1.0f / L2;
    }
  }
}

// ------- Kernel 2: WMMA rasterizer (1 wave = 16 pixels, async double-buffer) -------
__global__ void __launch_bounds__(32)
raster_kernel(const float* __restrict__ fsetup,
              const float* __restrict__ face_features,
              const float* __restrict__ texture,
              float* __restrict__ out,
              float* __restrict__ maskbuf) {
  __shared__ float lfs[2][16*FS];      // double buffer, 2 KB each
  const int lane = threadIdx.x;        // wave32
  const int n    = lane & 15;          // pixel slot (WMMA column)
  const bool hi  = lane >= 16;
  const int frame = blockIdx.y;
  const int tile  = blockIdx.x;
  const int pix   = tile*16 + n;
  const int y = pix >> 7, x = pix & 127;

  const float step = 2.0f / 127.0f;
  const float px = -1.0f + x*step;     // xs = linspace(-1,1,128)
  const float py =  1.0f - y*step;     // ys = linspace(1,-1,128)

  // B (4x16): K rows = {px, py, 1, 0}; lanes 0-15 hold K={0,1}, lanes 16-31 K={2,3}
  v2f Bm; Bm.x = hi ? 1.0f : px; Bm.y = hi ? 0.0f : py;
  const int asel = hi ? 2 : 0;         // branchless A fetch: {Cx,Cy} vs {Cc,0}

  const float* fsF = fsetup + (size_t)frame*NFACES*FS;
  // 32-bit LDS byte offsets of the two buffers for this lane's 64-byte slice
  unsigned ldsbase =
      (unsigned)(size_t)(__attribute__((address_space(3))) float*)(&lfs[0][0]);
  unsigned lds0 = ldsbase + (unsigned)lane*64u;
  unsigned lds1 = lds0 + 16u*FS*4u;

  float bestS = -10000000000.0f; int bestM = 0; float bw1 = 0.0f, bw2 = 0.0f;
  float prodNot = 1.0f; int anyIn = 0;

  // prologue: stream chunk 0 into buffer 0
  async_stage(fsF + lane*16, lds0);

  for (int c = 0; c < NCHUNK; c++) {
    const int cur = c & 1;
    const float* lbuf = &lfs[cur][0];
    if (c + 1 < NCHUNK) {
      // prefetch next chunk into the other buffer, then wait for current chunk:
      // async loads complete in order, so <=4 outstanding => current 4 are done
      async_stage(fsF + (c+1)*16*FS + lane*16, cur ? lds0 : lds1);
      asm volatile("s_wait_asynccnt 4" ::: "memory");
    } else {
      asm volatile("s_wait_asynccnt 0" ::: "memory");
    }

    // A (16x4): face m = lane%16; lanes 0-15 hold K={Cx,Cy}, lanes 16-31 K={Cc,0}
    const float* fA = lbuf + n*FS + asel;
    v2f A1 = *(const v2f*)(fA + 0);
    v2f A2 = *(const v2f*)(fA + 4);
    v2f AZ = *(const v2f*)(fA + 8);
    v8f cz = {0,0,0,0,0,0,0,0};
    v8f W1 = __builtin_amdgcn_wmma_f32_16x16x4_f32(false, A1, false, Bm, (short)0, cz, false, false);
    v8f W2 = __builtin_amdgcn_wmma_f32_16x16x4_f32(false, A2, false, Bm, (short)0, cz, false, false);
    v8f SC = __builtin_amdgcn_wmma_f32_16x16x4_f32(false, AZ, false, Bm, (short)0, cz, false, false);

    // D layout: lane owns pixel n, faces r + 8*hi in VGPR r
#pragma unroll
    for (int r = 0; r < 8; r++) {
      int mloc = r + (hi ? 8 : 0);
      int m = c*16 + mloc;
      const float* fm = lbuf + mloc*FS;
      float w1 = W1[r], w2 = W2[r], sc = SC[r];
      float w0 = 1.0f - w1 - w2;
      float validf = fm[12];
      bool inside = (w0 >= 0.0f) & (w1 >= 0.0f) & (w2 >= 0.0f) & (validf > 0.5f);
      anyIn |= inside ? 1 : 0;
      float score = inside ? sc : -10000000000.0f;
      if (score > bestS) { bestS = score; bestM = m; bw1 = w1; bw2 = w2; }
      if (validf > 0.5f) {               // invalid faces contribute prob=0
        float d2 = 3.4e38f;
#pragma unroll
        for (int e = 0; e < 3; e++) {
          const float* fe = fm + 13 + e*5;
          float rx = px - fe[0], ry = py - fe[1];
          float ex = fe[2], ey = fe[3], invL2 = fe[4];
          float tt = (rx*ex + ry*ey) * invL2;
          tt = fminf(fmaxf(tt, 0.0f), 1.0f);
          float dxe = rx - tt*ex, dye = ry - tt*ey;
          d2 = fminf(d2, dxe*dxe + dye*dye);
        }
        if (inside) d2 = 0.0f;
        float prob = expf(-SIGMAINV_C * d2);
        prodNot *= (1.0f - prob);
      }
    }
  }

  // combine the two lane halves (faces split 0-7 / 8-15 per chunk, same pixel)
  float oS = __shfl_xor(bestS, 16);
  int   oM = __shfl_xor(bestM, 16);
  float o1 = __shfl_xor(bw1, 16);
  float o2 = __shfl_xor(bw2, 16);
  float oP = __shfl_xor(prodNot, 16);
  int   oI = __shfl_xor(anyIn, 16);
  if (oS > bestS || (oS == bestS && oM < bestM)) { bestS = oS; bestM = oM; bw1 = o1; bw2 = o2; }
  prodNot *= oP;
  anyIn |= oI;

  if (!hi) {
    maskbuf[(size_t)frame*HH*WW + pix] = 1.0f - prodNot;
    float u = 0.0f, v = 0.0f;
    if (anyIn) {                                  // feats *= hit
      const float* ff = face_features + (size_t)bestM*6;   // (3 verts, 2 uv)
      float w0 = 1.0f - bw1 - bw2;
      u = w0*ff[0] + bw1*ff[2] + bw2*ff[4];
      v = w0*ff[1] + bw1*ff[3] + bw2*ff[5];
    }
    // bilinear texture sample (3,256,256)
    float uu = fminf(fmaxf(u, 0.0f), 1.0f) * 255.0f;
    float vv = (1.0f - fminf(fmaxf(v, 0.0f), 1.0f)) * 255.0f;
    float xf = floorf(uu), yf = floorf(vv);
    float fx = uu - xf, fy = vv - yf;
    int x0i = min(max((int)xf, 0), 255); int x1i = min(x0i + 1, 255);
    int y0i = min(max((int)yf, 0), 255); int y1i = min(y0i + 1, 255);
#pragma unroll
    for (int ch = 0; ch < 3; ch++) {
      const float* tc = texture + (size_t)ch*256*256;
      float t00 = tc[y0i*256+x0i], t01 = tc[y0i*256+x1i];
      float t10 = tc[y1i*256+x0i], t11 = tc[y1i*256+x1i];
      float val = t00*(1.0f-fx)*(1.0f-fy) + t01*fx*(1.0f-fy)
                + t10*(1.0f-fx)*fy        + t11*fx*fy;
      out[(((size_t)frame*4 + ch)*HH + y)*WW + x] = val;
    }
  }
}

// ---------------- Kernel 3: 5x5 min-pool erode -> channel 3 ----------------
__global__ void erode_kernel(const float* __restrict__ maskbuf, float* __restrict__ out) {
  int idx = blockIdx.x*blockDim.x + threadIdx.x;
  if (idx >= NFRAMES*HH*WW) return;
  int t = idx / (HH*WW);
  int rem = idx % (HH*WW);
  int y = rem >> 7, x = rem & 127;
  const float* mb = maskbuf + (size_t)t*HH*WW;
  float mn = 3.4e38f;
  for (int dy = -2; dy <= 2; dy++) {
    int yy = y + dy; if (yy < 0 || yy >= HH) continue;
    for (int dx = -2; dx <= 2; dx++) {
      int xx = x + dx; if (xx < 0 || xx >= WW) continue;
      mn = fminf(mn, mb[yy*WW + xx]);
    }
  }
  out[(((size_t)t*4 + 3)*HH + y)*WW + x] = mn;
}

extern "C" void kernel_launch(void* const* d_in, const int* in_sizes, int n_in,
                              void* d_out, int out_size, void* d_ws, size_t ws_size,
                              hipStream_t stream) {
  const float* vertices      = (const float*)d_in[0]; // (1,162,3)
  const float* translation   = (const float*)d_in[1]; // (1,1,1,2,3)
  const float* quaternion    = (const float*)d_in[2]; // (1,1,2,4)
  const float* expv          = (const float*)d_in[3]; // (1,)
  const float* face_features = (const float*)d_in[4]; // (1,320,3,2)
  const float* texture       = (const float*)d_in[5]; // (1,3,256,256)
  const int*   faces         = (const int*)d_in[6];   // (320,3)
  float* out = (float*)d_out;                         // (1,1,4,4,128,128)

  float* fsetup  = (float*)d_ws;                      // 4*320*32 floats
  float* maskbuf = fsetup + (size_t)NFRAMES*NFACES*FS;// 4*128*128 floats

  setup_kernel<<<NFRAMES, 256, 0, stream>>>(vertices, translation, quaternion,
                                            expv, faces, fsetup);
  dim3 rg(HH*WW/16, NFRAMES);
  raster_kernel<<<rg, 32, 0, stream>>>(fsetup, face_features, texture, out, maskbuf);
  erode_kernel<<<(NFRAMES*HH*WW + 255)/256, 256, 0, stream>>>(maskbuf, out);
}